// FusedSqueezeExcitation_55439437857264
// MI455X (gfx1250) — compile-verified
//
#include <hip/hip_runtime.h>
#include <hip/hip_bf16.h>

typedef __attribute__((ext_vector_type(2))) float v2f;
typedef __attribute__((ext_vector_type(4))) float f4;
typedef __attribute__((ext_vector_type(8))) float v8f;

#define BATCH   64
#define CH      256
#define CSQ     64
#define HW      3136
#define HW4     784
#define NPLANE  (BATCH * CH)   // 16384

// ---------------------------------------------------------------------------
// K1: per-(b,c) plane: mean + min + max of x  (memory-bound pass 1)
// ---------------------------------------------------------------------------
__global__ __launch_bounds__(256) void se_pool_minmax(
    const float* __restrict__ x,
    float* __restrict__ pooled, float* __restrict__ mn, float* __restrict__ mx) {
  const int plane = blockIdx.x;
  const f4* xp = (const f4*)(x + (size_t)plane * HW);
  float s = 0.0f, lo = 3.402823e38f, hi = -3.402823e38f;
  for (int i = threadIdx.x; i < HW4; i += 256) {
    f4 v = xp[i];
    s += (v.x + v.y) + (v.z + v.w);
    lo = fminf(lo, fminf(fminf(v.x, v.y), fminf(v.z, v.w)));
    hi = fmaxf(hi, fmaxf(fmaxf(v.x, v.y), fmaxf(v.z, v.w)));
  }
  // wave32 shuffle reduction
  for (int o = 16; o > 0; o >>= 1) {
    s  += __shfl_xor(s, o, 32);
    lo  = fminf(lo, __shfl_xor(lo, o, 32));
    hi  = fmaxf(hi, __shfl_xor(hi, o, 32));
  }
  __shared__ float ss[8], sl[8], sh[8];
  const int lane = threadIdx.x & 31, wv = threadIdx.x >> 5;
  if (lane == 0) { ss[wv] = s; sl[wv] = lo; sh[wv] = hi; }
  __syncthreads();
  if (threadIdx.x == 0) {
    float S = ss[0], L = sl[0], H = sh[0];
    for (int w = 1; w < 8; ++w) { S += ss[w]; L = fminf(L, sl[w]); H = fmaxf(H, sh[w]); }
    pooled[plane] = S * (1.0f / (float)HW);
    mn[plane] = L;
    mx[plane] = H;
  }
}

// ---------------------------------------------------------------------------
// K2: one WGP does the whole SE core with f32 WMMA, plus the scalar quant math.
//   GEMM1: h[64,64]  = relu(pooled[64,256] @ w1^T + b1)   (K=256)
//   GEMM2: g[64,256] = hardsigmoid(h @ w2^T + b2)         (K=64)
//   rmin = min g*minx, rmax = max g*maxx  -> EMA -> s, z
// Fragment layouts per CDNA5 ISA 7.12.2 (32-bit A 16x4, row-striped B/C/D).
// ---------------------------------------------------------------------------
__global__ __launch_bounds__(1024) void se_gemm_scale(
    const float* __restrict__ pooled,
    const float* __restrict__ w1, const float* __restrict__ b1,
    const float* __restrict__ w2, const float* __restrict__ b2,
    const float* __restrict__ mn, const float* __restrict__ mx,
    const float* __restrict__ act_range,
    float* __restrict__ g_out, float* __restrict__ sz) {
  __shared__ float h_lds[BATCH * CSQ];   // 16 KB
  __shared__ float rl[32], rh[32];

  const int tid  = threadIdx.x;
  const int lane = tid & 31;
  const int wv   = tid >> 5;
  const int l15  = lane & 15;
  const int kh   = (lane >> 4) << 1;   // K sub-offset: 0 (lanes 0-15) / 2 (lanes 16-31)
  const int rofs = (lane >> 4) << 3;   // C/D row offset: 0 / 8

  // ---- GEMM1: waves 0..15, one 16x16 output tile each, K = 256 ----
  if (wv < 16) {
    const int m0 = (wv >> 2) << 4;
    const int n0 = (wv & 3) << 4;
    v8f acc = {};
    const float* ap = pooled + (m0 + l15) * CH;   // A row (per lane)
    const float* bp = w1 + (n0 + l15) * CH;       // B col n = w1 row n (contiguous K)
    for (int k0 = 0; k0 < CH; k0 += 4) {
      v2f a = *(const v2f*)(ap + k0 + kh);
      v2f b = *(const v2f*)(bp + k0 + kh);
      acc = __builtin_amdgcn_wmma_f32_16x16x4_f32(false, a, false, b,
                                                  (short)0, acc, false, false);
    }
    const float bias = b1[n0 + l15];
    for (int r = 0; r < 8; ++r)
      h_lds[(m0 + r + rofs) * CSQ + (n0 + l15)] = fmaxf(acc[r] + bias, 0.0f);
  }
  __syncthreads();

  // ---- GEMM2: 64 tiles over 32 waves (2 each), K = 64 ----
  for (int t = wv; t < 64; t += 32) {
    const int m0 = (t >> 4) << 4;
    const int n0 = (t & 15) << 4;
    v8f acc = {};
    const float* ap = h_lds + (m0 + l15) * CSQ;
    const float* bp = w2 + (n0 + l15) * CSQ;      // B col c = w2 row c (contiguous K)
    for (int k0 = 0; k0 < CSQ; k0 += 4) {
      v2f a = *(const v2f*)(ap + k0 + kh);
      v2f b = *(const v2f*)(bp + k0 + kh);
      acc = __builtin_amdgcn_wmma_f32_16x16x4_f32(false, a, false, b,
                                                  (short)0, acc, false, false);
    }
    const int col  = n0 + l15;
    const float bias = b2[col];
    for (int r = 0; r < 8; ++r) {
      const int row = m0 + r + rofs;
      float v = acc[r] + bias;
      float gg = fminf(fmaxf(v * (1.0f / 6.0f) + 0.5f, 0.0f), 1.0f); // hardsigmoid
      g_out[row * CH + col] = gg;
    }
  }
  __threadfence_block();
  __syncthreads();

  // ---- rmin/rmax over all (b,c): out-plane extrema are g * {minx,maxx} (g>=0) ----
  float lo = 3.402823e38f, hi = -3.402823e38f;
  for (int i = tid; i < NPLANE; i += 1024) {
    float gg = g_out[i];
    lo = fminf(lo, gg * mn[i]);
    hi = fmaxf(hi, gg * mx[i]);
  }
  for (int o = 16; o > 0; o >>= 1) {
    lo = fminf(lo, __shfl_xor(lo, o, 32));
    hi = fmaxf(hi, __shfl_xor(hi, o, 32));
  }
  if (lane == 0) { rl[wv] = lo; rh[wv] = hi; }
  __syncthreads();
  if (tid == 0) {
    float L = rl[0], H = rh[0];
    for (int w = 1; w < 32; ++w) { L = fminf(L, rl[w]); H = fmaxf(H, rh[w]); }
    const float nmin = act_range[0] * 0.999f + L * 0.001f;
    const float nmax = act_range[1] * 0.999f + H * 0.001f;
    const float s = (nmax - nmin) * (1.0f / 255.0f);
    const float z = -rintf(nmin / s);             // jnp.round == round-half-even
    sz[0] = s;
    sz[1] = z;
  }
}

// ---------------------------------------------------------------------------
// K3: out = fakequant(x * g)  (memory-bound pass 2; NT stores keep x in L2)
// ---------------------------------------------------------------------------
__global__ __launch_bounds__(256) void se_apply_quant(
    const float* __restrict__ x, const float* __restrict__ g,
    const float* __restrict__ sz, float* __restrict__ out) {
  const int plane = blockIdx.x;
  const float gg = g[plane];
  const float s = sz[0], z = sz[1];
  const float inv_s = 1.0f / s;
  const f4* xp = (const f4*)(x + (size_t)plane * HW);
  f4* op = (f4*)(out + (size_t)plane * HW);
  for (int i = threadIdx.x; i < HW4; i += 256) {
    f4 v = xp[i];
    f4 q;
    q.x = (fminf(fmaxf(rintf(fmaf(v.x * gg, inv_s, z)), 0.0f), 255.0f) - z) * s;
    q.y = (fminf(fmaxf(rintf(fmaf(v.y * gg, inv_s, z)), 0.0f), 255.0f) - z) * s;
    q.z = (fminf(fmaxf(rintf(fmaf(v.z * gg, inv_s, z)), 0.0f), 255.0f) - z) * s;
    q.w = (fminf(fmaxf(rintf(fmaf(v.w * gg, inv_s, z)), 0.0f), 255.0f) - z) * s;
    __builtin_nontemporal_store(q, op + i);
  }
}

// ---------------------------------------------------------------------------
extern "C" void kernel_launch(void* const* d_in, const int* in_sizes, int n_in,
                              void* d_out, int out_size, void* d_ws, size_t ws_size,
                              hipStream_t stream) {
  const float* x   = (const float*)d_in[0];
  const float* w1  = (const float*)d_in[1];
  const float* b1  = (const float*)d_in[2];
  const float* w2  = (const float*)d_in[3];
  const float* b2  = (const float*)d_in[4];
  const float* act = (const float*)d_in[5];
  float* out = (float*)d_out;

  // workspace layout (floats): pooled | minx | maxx | g | sz   (~256 KB)
  float* ws     = (float*)d_ws;
  float* pooled = ws;
  float* mn     = ws + NPLANE;
  float* mx     = ws + 2 * NPLANE;
  float* g      = ws + 3 * NPLANE;
  float* sz     = ws + 4 * NPLANE;

  se_pool_minmax<<<NPLANE, 256, 0, stream>>>(x, pooled, mn, mx);
  se_gemm_scale<<<1, 1024, 0, stream>>>(pooled, w1, b1, w2, b2, mn, mx, act, g, sz);
  se_apply_quant<<<NPLANE, 256, 0, stream>>>(x, g, sz, out);
}